// CustomMultiHeadAttention_79860621902019
// MI455X (gfx1250) — compile-verified
//
#include <hip/hip_runtime.h>
#include <hip/hip_bf16.h>

// Shapes from the reference
#define BB 8
#define QLEN 2048
#define DD 1024
#define HH 16
#define KVN 256
#define HD 64

typedef __attribute__((ext_vector_type(2))) float v2f;
typedef __attribute__((ext_vector_type(8))) float v8f;

// ---------------------------------------------------------------------------
// Kernel 1: K/V projection via V_WMMA_F32_16X16X4_F32.
// Kvec[b,d] = sum_c key[b,c] * k_weight[d,c]   (B=8 rows padded to 16)
// One wave per 16-wide d tile; blockIdx 0..63 -> K proj, 64..127 -> V proj.
// A frag (16x4 f32): lanes 0-15 M=lane K={0,1}; lanes 16-31 M=lane-16 K={2,3}
// B frag (4x16 f32): lanes 0-15 N=lane K={0,1}; lanes 16-31 N=lane-16 K={2,3}
// D frag (16x16 f32): lane group selects N=lane&15, VGPR j holds M=j(+8 hi).
// ---------------------------------------------------------------------------
__global__ __launch_bounds__(32) void proj_wmma_kernel(
    const float* __restrict__ key, const float* __restrict__ value,
    const float* __restrict__ wk, const float* __restrict__ wv,
    float* __restrict__ kvec, float* __restrict__ vvec) {
  const int tile = blockIdx.x & 63;
  const bool isV = blockIdx.x >= 64;
  const float* __restrict__ src = isV ? value : key;   // (8, 1024)
  const float* __restrict__ w   = isV ? wv : wk;       // (1024 d, 1024 c)
  float* __restrict__ out       = isV ? vvec : kvec;   // (8, 1024)

  const int lane = threadIdx.x;
  const int m    = lane & 15;          // A row (batch), valid if < 8
  const int kg   = (lane >> 4) << 1;   // 0 or 2 : K sub-pair
  const int d0   = tile * 16;

  // Zero-pad batch rows 8..15 without divergent control flow.
  const float sel = (m < BB) ? 1.0f : 0.0f;
  const float* __restrict__ arow = src + (m & 7) * DD;
  const float* __restrict__ brow = w + (d0 + m) * DD;  // N = lane&15 == m

  v8f acc = {};
  for (int k0 = 0; k0 < DD; k0 += 4) {
    v2f a, b;
    float2 av = *(const float2*)(arow + k0 + kg);
    float2 bv = *(const float2*)(brow + k0 + kg);
    a.x = av.x * sel; a.y = av.y * sel;
    b.x = bv.x;       b.y = bv.y;
    acc = __builtin_amdgcn_wmma_f32_16x16x4_f32(
        /*neg_a=*/false, a, /*neg_b=*/false, b,
        /*c_mod=*/(short)0, acc, /*reuse_a=*/false, /*reuse_b=*/false);
  }
  if (lane < 16) {  // lanes 0-15 hold M=0..7 (all real batch rows)
#pragma unroll
    for (int j = 0; j < 8; ++j) out[j * DD + d0 + lane] = acc[j];
  }
}

// ---------------------------------------------------------------------------
// Kernel 2: per-batch scalars. scal[0..7]=clip@w2, [8..15]=cmax, [16..23]=cmin,
// [24]=sum(w1). Wave w handles batch b=w.
// ---------------------------------------------------------------------------
__global__ __launch_bounds__(256) void scalars_kernel(
    const float* __restrict__ clip, const float* __restrict__ sw,
    float* __restrict__ scal) {
  const int b = threadIdx.x >> 5, lane = threadIdx.x & 31;
  float cs = 0.f, cmx = -1e30f, cmn = 1e30f, wsum = 0.f;
  for (int k = lane; k < KVN; k += 32) {
    float c = clip[b * KVN + k];
    cs = fmaf(c, sw[KVN + k], cs);
    cmx = fmaxf(cmx, c);
    cmn = fminf(cmn, c);
    wsum += sw[k];
  }
  for (int off = 16; off; off >>= 1) {
    cs += __shfl_xor(cs, off, 32);
    cmx = fmaxf(cmx, __shfl_xor(cmx, off, 32));
    cmn = fminf(cmn, __shfl_xor(cmn, off, 32));
    wsum += __shfl_xor(wsum, off, 32);
  }
  if (lane == 0) {
    scal[b] = cs; scal[8 + b] = cmx; scal[16 + b] = cmn;
    if (b == 0) scal[24] = wsum;
  }
}

// ---------------------------------------------------------------------------
// Kernel 3: one wave per (b,h,q). s0 = Q·Kvec/8 (64-wide dot), then
// attn[b,h,q,k] = softmax_k(L * clip[b,k]) with L = s0*Σw1 + cs2[b] + scale_b.
// Stable max = max(L*cmax, L*cmin) covers both signs of L.
// ---------------------------------------------------------------------------
__global__ __launch_bounds__(256) void attn_softmax_kernel(
    const float* __restrict__ q, const float* __restrict__ clip,
    const float* __restrict__ kvec, const float* __restrict__ scal,
    const float* __restrict__ sb, float* __restrict__ attn) {
  const int warp = threadIdx.x >> 5;
  const int lane = threadIdx.x & 31;
  const long idx = (long)blockIdx.x * 8 + warp;  // flat (b*H + h)*Q + q
  const int qpos = (int)(idx & (QLEN - 1));
  const int bh   = (int)(idx >> 11);
  const int h    = bh & (HH - 1);
  const int b    = bh >> 4;

  // 64-element dot product, 2 elems per lane
  const float* __restrict__ qrow =
      q + ((long)(b * QLEN + qpos) * DD) + h * HD + lane * 2;
  const float* __restrict__ krow = kvec + b * DD + h * HD + lane * 2;
  float2 qa = *(const float2*)qrow;
  float2 ka = *(const float2*)krow;
  float dot = fmaf(qa.x, ka.x, qa.y * ka.y);
  for (int off = 16; off; off >>= 1) dot += __shfl_xor(dot, off, 32);
  const float s0 = dot * 0.125f;                 // 1/sqrt(HD)

  const float L = fmaf(s0, scal[24], scal[b] + sb[0]);
  const float m = fmaxf(L * scal[8 + b], L * scal[16 + b]);

  const float4* __restrict__ cp = (const float4*)(clip + b * KVN + lane * 8);
  float4 c0 = cp[0], c1 = cp[1];
  float4 e0, e1;
  e0.x = __expf(fmaf(L, c0.x, -m));
  e0.y = __expf(fmaf(L, c0.y, -m));
  e0.z = __expf(fmaf(L, c0.z, -m));
  e0.w = __expf(fmaf(L, c0.w, -m));
  e1.x = __expf(fmaf(L, c1.x, -m));
  e1.y = __expf(fmaf(L, c1.y, -m));
  e1.z = __expf(fmaf(L, c1.z, -m));
  e1.w = __expf(fmaf(L, c1.w, -m));
  float s = (e0.x + e0.y) + (e0.z + e0.w) + (e1.x + e1.y) + (e1.z + e1.w);
  for (int off = 16; off; off >>= 1) s += __shfl_xor(s, off, 32);
  const float inv = 1.0f / s;
  e0.x *= inv; e0.y *= inv; e0.z *= inv; e0.w *= inv;
  e1.x *= inv; e1.y *= inv; e1.z *= inv; e1.w *= inv;

  float4* __restrict__ op = (float4*)(attn + idx * KVN + lane * 8);
  op[0] = e0;
  op[1] = e1;
}

// ---------------------------------------------------------------------------
// Kernel 4: out[b,q,:] = Vvec[b,:] (rank-1 V + softmax-sums-to-1 => broadcast)
// ---------------------------------------------------------------------------
__global__ __launch_bounds__(256) void out_bcast_kernel(
    const float* __restrict__ vvec, float* __restrict__ out) {
  const int row = blockIdx.x;            // b*QLEN + q
  const int b = row >> 11;
  float4 v = *(const float4*)(vvec + b * DD + threadIdx.x * 4);
  *(float4*)(out + (long)row * DD + threadIdx.x * 4) = v;
}

extern "C" void kernel_launch(void* const* d_in, const int* in_sizes, int n_in,
                              void* d_out, int out_size, void* d_ws,
                              size_t ws_size, hipStream_t stream) {
  const float* q     = (const float*)d_in[0];
  const float* key   = (const float*)d_in[1];
  const float* value = (const float*)d_in[2];
  const float* clip  = (const float*)d_in[3];
  const float* wk    = (const float*)d_in[4];
  const float* wv    = (const float*)d_in[5];
  const float* sw    = (const float*)d_in[6];
  const float* sb    = (const float*)d_in[7];

  float* out  = (float*)d_out;                       // (B, QLEN, D)
  float* attn = out + (long)BB * QLEN * DD;          // (B, H, QLEN, KV)

  float* ws   = (float*)d_ws;
  float* kvec = ws;            // 8192 floats
  float* vvec = ws + 8192;     // 8192 floats
  float* scal = ws + 16384;    // 25 floats

  proj_wmma_kernel<<<128, 32, 0, stream>>>(key, value, wk, wv, kvec, vvec);
  scalars_kernel<<<1, 256, 0, stream>>>(clip, sw, scal);
  attn_softmax_kernel<<<(BB * HH * QLEN) / 8, 256, 0, stream>>>(
      q, clip, kvec, scal, sb, attn);
  out_bcast_kernel<<<BB * QLEN, 256, 0, stream>>>(vvec, out);
}